// SS2D_26465588478577
// MI455X (gfx1250) — compile-verified
//
#include <hip/hip_runtime.h>
#include <hip/hip_bf16.h>

// ---------------------------------------------------------------------------
// SS2D (VMamba) block for MI455X (gfx1250, wave32).
// GEMMs use V_WMMA_F32_16X16X4_F32 with 2x4 register blocking
// (32x64 tile per wave) and software-pipelined (double-buffered) fragment
// loads so next k-step loads overlap current k-step WMMAs.
// Selective scan uses a 32-way chunked linear-recurrence scan.
// ---------------------------------------------------------------------------

typedef __attribute__((ext_vector_type(2))) float v2f;
typedef __attribute__((ext_vector_type(8))) float v8f;

#define D_MODEL 96
#define D_INNER 192
#define D_STATE 16
#define DT_RANK 6
#define N_DBC   38          // DT_RANK + 2*D_STATE
#define SEQ_L   4096
#define N_SEQ   8           // 4 directions * batch 2
#define M_ALL   (N_SEQ * SEQ_L)   // 32768
#define HW      64
#define BATCH   2
#define NCHUNK  32
#define CLEN    128         // SEQ_L / NCHUNK
#define LOG2E   1.44269504088896340736f

// ---------------------------------------------------------------------------
// fp32 WMMA GEMM:  C[M x N] = A[M x K] * W[N x K]^T
// One wave computes a 32x64 tile (2 m-tiles x 4 n-tiles); 8 waves per block.
// A fragment: lane = (row = lane&15, kbase = (lane>>4)*2) -> float2 {k, k+1}
// B fragment: lane = (col = lane&15, kbase = (lane>>4)*2) -> float2 of W row
// D layout  : vgpr v -> row m0 + (lane>>4)*8 + v, col = n0 + (lane&15)
// Requires M % 32 == 0 (true for all call sites) and K % 4 == 0.
// ---------------------------------------------------------------------------
__global__ __launch_bounds__(256) void k_gemm_f32(
    const float* __restrict__ A, const float* __restrict__ W,
    float* __restrict__ C, int M, int N, int K,
    int lda, int ldb, int ldc) {
  const int lane = threadIdx.x & 31;
  const int wave = threadIdx.x >> 5;
  const int Mt = M >> 5;                       // 32-row supertiles
  const int Ng = (((N + 15) >> 4) + 3) >> 2;   // groups of 4 n-tiles
  const int tile = blockIdx.x * 8 + wave;
  if (tile >= Mt * Ng) return;                 // wave-uniform: EXEC stays full
  const int g  = tile % Ng;
  const int mt = tile / Ng;
  const int m0 = mt << 5, n0 = g << 6;

  const int row  = lane & 15;
  const int koff = (lane >> 4) << 1;

  const float* ap0 = A + (size_t)(m0 + row) * lda + koff;
  const float* ap1 = ap0 + (size_t)16 * lda;

  const float* bp[4];
  bool nok[4];
#pragma unroll
  for (int j = 0; j < 4; ++j) {
    int nc = n0 + j * 16 + row;
    nok[j] = (nc < N);
    bp[j] = W + (size_t)(nok[j] ? nc : 0) * ldb + koff;
  }

  v8f acc[8];
#pragma unroll
  for (int j = 0; j < 8; ++j) acc[j] = (v8f){};

  // Software-pipelined fragment loads: prologue for k = 0.
  v2f a0 = *(const v2f*)(ap0);
  v2f a1 = *(const v2f*)(ap1);
  v2f b0 = *(const v2f*)(bp[0]);
  v2f b1 = *(const v2f*)(bp[1]);
  v2f b2 = *(const v2f*)(bp[2]);
  v2f b3 = *(const v2f*)(bp[3]);

  for (int k = 4; k <= K; k += 4) {
    v2f na0, na1, nb0, nb1, nb2, nb3;
    if (k < K) {                                // wave-uniform branch
      na0 = *(const v2f*)(ap0 + k);
      na1 = *(const v2f*)(ap1 + k);
      nb0 = *(const v2f*)(bp[0] + k);
      nb1 = *(const v2f*)(bp[1] + k);
      nb2 = *(const v2f*)(bp[2] + k);
      nb3 = *(const v2f*)(bp[3] + k);
    }
    acc[0] = __builtin_amdgcn_wmma_f32_16x16x4_f32(false, a0, false, b0, (short)0, acc[0], false, false);
    acc[4] = __builtin_amdgcn_wmma_f32_16x16x4_f32(false, a1, false, b0, (short)0, acc[4], false, false);
    acc[1] = __builtin_amdgcn_wmma_f32_16x16x4_f32(false, a0, false, b1, (short)0, acc[1], false, false);
    acc[5] = __builtin_amdgcn_wmma_f32_16x16x4_f32(false, a1, false, b1, (short)0, acc[5], false, false);
    acc[2] = __builtin_amdgcn_wmma_f32_16x16x4_f32(false, a0, false, b2, (short)0, acc[2], false, false);
    acc[6] = __builtin_amdgcn_wmma_f32_16x16x4_f32(false, a1, false, b2, (short)0, acc[6], false, false);
    acc[3] = __builtin_amdgcn_wmma_f32_16x16x4_f32(false, a0, false, b3, (short)0, acc[3], false, false);
    acc[7] = __builtin_amdgcn_wmma_f32_16x16x4_f32(false, a1, false, b3, (short)0, acc[7], false, false);
    a0 = na0; a1 = na1; b0 = nb0; b1 = nb1; b2 = nb2; b3 = nb3;
  }

  const int rbase = (lane >> 4) << 3;          // 0 or 8
#pragma unroll
  for (int j = 0; j < 4; ++j) {
    if (!nok[j]) continue;
    const int ncol = n0 + j * 16 + row;
    float* cp0 = C + (size_t)(m0 + rbase) * ldc + ncol;
    float* cp1 = C + (size_t)(m0 + 16 + rbase) * ldc + ncol;
#pragma unroll
    for (int v = 0; v < 8; ++v) {
      cp0[(size_t)v * ldc] = acc[j][v];
      cp1[(size_t)v * ldc] = acc[4 + j][v];
    }
  }
}

// ---------------------------------------------------------------------------
// Build xs[m, c]: 4 directional re-orderings of x (b, c, h, w)
// n = m>>12 : n = dir*2 + b
// ---------------------------------------------------------------------------
__global__ __launch_bounds__(256) void k_gather_dirs(
    const float* __restrict__ x, float* __restrict__ xs) {
  int idx = blockIdx.x * 256 + threadIdx.x;
  if (idx >= M_ALL * D_MODEL) return;
  int c = idx % D_MODEL;
  int m = idx / D_MODEL;
  int n = m >> 12;
  int l = m & (SEQ_L - 1);
  int b = n & 1, di = n >> 1;
  int h, w;
  if (di == 0)      { h = l >> 6;        w = l & 63; }
  else if (di == 1) { h = l & 63;        w = l >> 6; }
  else if (di == 2) { h = l >> 6;        w = 63 - (l & 63); }
  else              { h = 63 - (l >> 6); w = l & 63; }
  xs[idx] = x[(((size_t)b * D_MODEL + c) << 12) + (h << 6) + w];
}

// ---------------------------------------------------------------------------
// Causal depthwise conv (k=4) over l + SiLU.  Reads xc = xz[:, 0:192].
// ---------------------------------------------------------------------------
__global__ __launch_bounds__(256) void k_conv_silu(
    const float* __restrict__ xz, const float* __restrict__ conv_w,
    const float* __restrict__ conv_b, float* __restrict__ xcv) {
  int idx = blockIdx.x * 256 + threadIdx.x;
  if (idx >= M_ALL * D_INNER) return;
  int d = idx % D_INNER;
  int m = idx / D_INNER;
  int l = m & (SEQ_L - 1);
  const float* base = xz + (size_t)m * (2 * D_INNER) + d;
  float acc = conv_b[d];
#pragma unroll
  for (int j = 0; j < 4; ++j) {
    int ls = l - 3 + j;
    float v = (ls >= 0) ? base[(long)(j - 3) * (2 * D_INNER)] : 0.f;
    acc = fmaf(conv_w[d * 4 + j], v, acc);
  }
  xcv[idx] = acc / (1.f + __expf(-acc));   // SiLU
}

// ---------------------------------------------------------------------------
// dt[m,d] = softplus( dbc[m, 0:6] . W_dt[d, :] + b_dt[d] )
// ---------------------------------------------------------------------------
__global__ __launch_bounds__(256) void k_dt(
    const float* __restrict__ dbc, const float* __restrict__ W_dt,
    const float* __restrict__ b_dt, float* __restrict__ dt) {
  int idx = blockIdx.x * 256 + threadIdx.x;
  if (idx >= M_ALL * D_INNER) return;
  int d = idx % D_INNER;
  int m = idx / D_INNER;
  float a = b_dt[d];
#pragma unroll
  for (int r = 0; r < DT_RANK; ++r)
    a = fmaf(dbc[(size_t)m * N_DBC + r], W_dt[d * DT_RANK + r], a);
  dt[idx] = (a > 20.f) ? a : log1pf(__expf(a));
}

// ---------------------------------------------------------------------------
// Chunked selective scan, pass 1: per (n, chunk, d) compute P = prod(dA),
// Q = accumulated input, for the 16-wide state.
// ---------------------------------------------------------------------------
__global__ __launch_bounds__(192) void k_scan1(
    const float* __restrict__ dbc, const float* __restrict__ dtb,
    const float* __restrict__ xcv, const float* __restrict__ A_log,
    float* __restrict__ Pb, float* __restrict__ Qb) {
  __shared__ float sB[CLEN][D_STATE];
  const int n = blockIdx.x >> 5;
  const int c = blockIdx.x & (NCHUNK - 1);
  const int d = threadIdx.x;
  const int l0 = c * CLEN;
  for (int i = threadIdx.x; i < CLEN * D_STATE; i += 192) {
    int li = i >> 4, si = i & 15;
    sB[li][si] = dbc[((size_t)(n * SEQ_L + l0 + li)) * N_DBC + DT_RANK + si];
  }
  __syncthreads();
  float A2[D_STATE];
#pragma unroll
  for (int s = 0; s < D_STATE; ++s)
    A2[s] = -__expf(A_log[d * D_STATE + s]) * LOG2E;
  float P[D_STATE], Q[D_STATE];
#pragma unroll
  for (int s = 0; s < D_STATE; ++s) { P[s] = 1.f; Q[s] = 0.f; }
  for (int li = 0; li < CLEN; ++li) {
    size_t m = (size_t)n * SEQ_L + l0 + li;
    float dt = dtb[m * D_INNER + d];
    float u  = dt * xcv[m * D_INNER + d];
#pragma unroll
    for (int s = 0; s < D_STATE; ++s) {
      float a = exp2f(dt * A2[s]);
      P[s] *= a;
      Q[s] = fmaf(a, Q[s], u * sB[li][s]);
    }
  }
  size_t o = ((size_t)(n * NCHUNK + c) * D_INNER + d) * D_STATE;
#pragma unroll
  for (int s = 0; s < D_STATE; ++s) { Pb[o + s] = P[s]; Qb[o + s] = Q[s]; }
}

// ---------------------------------------------------------------------------
// Chunk prefix fix: h_start(c) = P(c-1)*h_start(c-1) + Q(c-1), h_start(0)=0
// ---------------------------------------------------------------------------
__global__ __launch_bounds__(256) void k_scanfix(
    const float* __restrict__ Pb, const float* __restrict__ Qb,
    float* __restrict__ Hst) {
  int t = blockIdx.x * 256 + threadIdx.x;
  if (t >= N_SEQ * D_INNER) return;
  int n = t / D_INNER, d = t % D_INNER;
  float h[D_STATE];
#pragma unroll
  for (int s = 0; s < D_STATE; ++s) h[s] = 0.f;
  for (int c = 0; c < NCHUNK; ++c) {
    size_t o = ((size_t)(n * NCHUNK + c) * D_INNER + d) * D_STATE;
#pragma unroll
    for (int s = 0; s < D_STATE; ++s) {
      Hst[o + s] = h[s];
      h[s] = fmaf(Pb[o + s], h[s], Qb[o + s]);
    }
  }
}

// ---------------------------------------------------------------------------
// Pass 2: replay with correct h0, produce y = (sum_s h*C) + xc*Dp, gate by
// silu(z), store yact.
// ---------------------------------------------------------------------------
__global__ __launch_bounds__(192) void k_scan2(
    const float* __restrict__ dbc, const float* __restrict__ dtb,
    const float* __restrict__ xcv, const float* __restrict__ xz,
    const float* __restrict__ A_log, const float* __restrict__ Dp,
    const float* __restrict__ Hst, float* __restrict__ yact) {
  __shared__ float sB[CLEN][D_STATE];
  __shared__ float sC[CLEN][D_STATE];
  const int n = blockIdx.x >> 5;
  const int c = blockIdx.x & (NCHUNK - 1);
  const int d = threadIdx.x;
  const int l0 = c * CLEN;
  for (int i = threadIdx.x; i < CLEN * D_STATE; i += 192) {
    int li = i >> 4, si = i & 15;
    size_t rb = ((size_t)(n * SEQ_L + l0 + li)) * N_DBC + DT_RANK;
    sB[li][si] = dbc[rb + si];
    sC[li][si] = dbc[rb + D_STATE + si];
  }
  __syncthreads();
  float A2[D_STATE];
#pragma unroll
  for (int s = 0; s < D_STATE; ++s)
    A2[s] = -__expf(A_log[d * D_STATE + s]) * LOG2E;
  float h[D_STATE];
  {
    size_t o = ((size_t)(n * NCHUNK + c) * D_INNER + d) * D_STATE;
#pragma unroll
    for (int s = 0; s < D_STATE; ++s) h[s] = Hst[o + s];
  }
  const float dp = Dp[d];
  for (int li = 0; li < CLEN; ++li) {
    size_t m = (size_t)n * SEQ_L + l0 + li;
    float dt = dtb[m * D_INNER + d];
    float xc = xcv[m * D_INNER + d];
    float u  = dt * xc;
    float y  = 0.f;
#pragma unroll
    for (int s = 0; s < D_STATE; ++s) {
      float a = exp2f(dt * A2[s]);
      h[s] = fmaf(a, h[s], u * sB[li][s]);
      y = fmaf(h[s], sC[li][s], y);
    }
    y = fmaf(xc, dp, y);
    float z = xz[m * (2 * D_INNER) + D_INNER + d];
    yact[m * D_INNER + d] = y * (z / (1.f + __expf(-z)));
  }
}

// ---------------------------------------------------------------------------
// Merge 4 directions back into (b, l) x channel layout.
// ---------------------------------------------------------------------------
__global__ __launch_bounds__(256) void k_merge(
    const float* __restrict__ ys, float* __restrict__ ysum) {
  int idx = blockIdx.x * 256 + threadIdx.x;
  if (idx >= BATCH * SEQ_L * D_MODEL) return;
  int c = idx % D_MODEL;
  int rest = idx / D_MODEL;
  int l = rest & (SEQ_L - 1);
  int b = rest >> 12;
  int h = l >> 6, w = l & 63;
  float v =
      ys[((size_t)(0 + b) * SEQ_L + l) * D_MODEL + c] +
      ys[((size_t)(2 + b) * SEQ_L + (w * HW + h)) * D_MODEL + c] +
      ys[((size_t)(4 + b) * SEQ_L + (h * HW + (63 - w))) * D_MODEL + c] +
      ys[((size_t)(6 + b) * SEQ_L + ((63 - h) * HW + w)) * D_MODEL + c];
  ysum[idx] = v;
}

// ---------------------------------------------------------------------------
// Final: transpose (b,l,o) -> (b,o,h,w) and add bias.
// ---------------------------------------------------------------------------
__global__ __launch_bounds__(256) void k_final(
    const float* __restrict__ projres, const float* __restrict__ proj_b,
    float* __restrict__ out) {
  int idx = blockIdx.x * 256 + threadIdx.x;
  if (idx >= BATCH * D_MODEL * SEQ_L) return;
  int l = idx & (SEQ_L - 1);
  int o = (idx >> 12) % D_MODEL;
  int b = idx / (D_MODEL * SEQ_L);
  out[idx] = projres[((size_t)(b * SEQ_L + l)) * D_MODEL + o] + proj_b[o];
}

// ---------------------------------------------------------------------------
extern "C" void kernel_launch(void* const* d_in, const int* in_sizes, int n_in,
                              void* d_out, int out_size, void* d_ws, size_t ws_size,
                              hipStream_t stream) {
  const float* x      = (const float*)d_in[0];
  const float* W_in   = (const float*)d_in[1];
  const float* conv_w = (const float*)d_in[2];
  const float* conv_b = (const float*)d_in[3];
  const float* W_x    = (const float*)d_in[4];
  const float* W_dt   = (const float*)d_in[5];
  const float* b_dt   = (const float*)d_in[6];
  const float* A_log  = (const float*)d_in[7];
  const float* D_par  = (const float*)d_in[8];
  const float* W_out  = (const float*)d_in[9];
  const float* proj_w = (const float*)d_in[10];
  const float* proj_b = (const float*)d_in[11];
  float* out = (float*)d_out;

  // Workspace layout (floats)
  float* ws = (float*)d_ws;
  size_t off = 0;
  float* xs      = ws + off; off += (size_t)M_ALL * D_MODEL;       // 3.1M
  float* xz      = ws + off; off += (size_t)M_ALL * 2 * D_INNER;   // 12.6M
  float* xcv     = ws + off; off += (size_t)M_ALL * D_INNER;       // 6.3M
  float* dbc     = ws + off; off += (size_t)M_ALL * N_DBC;         // 1.25M
  float* dtb     = ws + off; off += (size_t)M_ALL * D_INNER;       // 6.3M
  float* yact    = ws + off; off += (size_t)M_ALL * D_INNER;       // 6.3M
  float* ys      = ws + off; off += (size_t)M_ALL * D_MODEL;       // 3.1M
  float* ysum    = ws + off; off += (size_t)BATCH * SEQ_L * D_MODEL;
  float* projres = ws + off; off += (size_t)BATCH * SEQ_L * D_MODEL;
  float* Pb      = ws + off; off += (size_t)N_SEQ * NCHUNK * D_INNER * D_STATE;
  float* Qb      = ws + off; off += (size_t)N_SEQ * NCHUNK * D_INNER * D_STATE;
  float* Hst     = ws + off; off += (size_t)N_SEQ * NCHUNK * D_INNER * D_STATE;

  auto gemm_blocks = [](int M, int N) {
    int Mt = M >> 5;
    int Ng = (((N + 15) >> 4) + 3) >> 2;
    return (Mt * Ng + 7) / 8;
  };

  // 1. directional gather
  k_gather_dirs<<<(M_ALL * D_MODEL + 255) / 256, 256, 0, stream>>>(x, xs);
  // 2. input projection: xz = xs @ W_in^T   (32768 x 384, K=96)
  k_gemm_f32<<<gemm_blocks(M_ALL, 2 * D_INNER), 256, 0, stream>>>(
      xs, W_in, xz, M_ALL, 2 * D_INNER, D_MODEL, D_MODEL, D_MODEL, 2 * D_INNER);
  // 3. causal depthwise conv + silu
  k_conv_silu<<<(M_ALL * D_INNER + 255) / 256, 256, 0, stream>>>(
      xz, conv_w, conv_b, xcv);
  // 4. dbc = xcv @ W_x^T  (32768 x 38, K=192)
  k_gemm_f32<<<gemm_blocks(M_ALL, N_DBC), 256, 0, stream>>>(
      xcv, W_x, dbc, M_ALL, N_DBC, D_INNER, D_INNER, D_INNER, N_DBC);
  // 5. dt = softplus(dbc[:, :6] @ W_dt^T + b_dt)
  k_dt<<<(M_ALL * D_INNER + 255) / 256, 256, 0, stream>>>(dbc, W_dt, b_dt, dtb);
  // 6. chunked selective scan
  k_scan1<<<N_SEQ * NCHUNK, 192, 0, stream>>>(dbc, dtb, xcv, A_log, Pb, Qb);
  k_scanfix<<<(N_SEQ * D_INNER + 255) / 256, 256, 0, stream>>>(Pb, Qb, Hst);
  k_scan2<<<N_SEQ * NCHUNK, 192, 0, stream>>>(
      dbc, dtb, xcv, xz, A_log, D_par, Hst, yact);
  // 7. out projection: ys = yact @ W_out^T  (32768 x 96, K=192)
  k_gemm_f32<<<gemm_blocks(M_ALL, D_MODEL), 256, 0, stream>>>(
      yact, W_out, ys, M_ALL, D_MODEL, D_INNER, D_INNER, D_INNER, D_MODEL);
  // 8. merge 4 directions
  k_merge<<<(BATCH * SEQ_L * D_MODEL + 255) / 256, 256, 0, stream>>>(ys, ysum);
  // 9. final projection: projres = ysum @ proj_w^T  (8192 x 96, K=96)
  k_gemm_f32<<<gemm_blocks(BATCH * SEQ_L, D_MODEL), 256, 0, stream>>>(
      ysum, proj_w, projres, BATCH * SEQ_L, D_MODEL, D_MODEL,
      D_MODEL, D_MODEL, D_MODEL);
  // 10. layout + bias
  k_final<<<(BATCH * D_MODEL * SEQ_L + 255) / 256, 256, 0, stream>>>(
      projres, proj_b, out);
}